// LightGCN_45509473468512
// MI455X (gfx1250) — compile-verified
//
#include <hip/hip_runtime.h>
#include <hip/hip_bf16.h>
#include <stdint.h>

#define NUM_USERS 200000
#define NUM_ITEMS 100000
#define NTOT      (NUM_USERS + NUM_ITEMS)
#define DIM       64
#define NUM_LAYERS 3

typedef unsigned int u32x4 __attribute__((ext_vector_type(4)));
typedef int          i32x8 __attribute__((ext_vector_type(8)));
typedef int          i32x4 __attribute__((ext_vector_type(4)));

#if __has_builtin(__builtin_amdgcn_tensor_load_to_lds) && __has_builtin(__builtin_amdgcn_s_wait_tensorcnt)
#define HAS_TDM 1
#else
#define HAS_TDM 0
#endif

#if HAS_TDM
// Build a 1-D TDM descriptor (D#) per CDNA5 ISA ch.8 and issue TENSOR_LOAD_TO_LDS.
// Copies `ndw` dwords from global `gaddr` to LDS byte offset `lds_off`.
// 6-arg builtin on this toolchain:
//   (uint32x4 g0, int32x8 g1, int32x4 g2, int32x4 g3, int32x8 extra, i32 cpol)
__device__ __forceinline__ void tdm_load_1d(uint32_t lds_off, const void* gaddr, uint32_t ndw) {
    uint64_t ga = (uint64_t)(uintptr_t)gaddr;
    u32x4 g0;
    g0[0] = 1u;                                                // count=1, user mode, no gather
    g0[1] = lds_off;                                           // lds_addr (bytes)
    g0[2] = (uint32_t)ga;                                      // global_addr[31:0]
    g0[3] = (uint32_t)((ga >> 32) & 0x01FFFFFFu) | (2u << 30); // global_addr[56:32] | type=2
    i32x8 g1;
    g1[0] = (int)(2u << 16);                                   // workgroup_mask=0, data_size=2 (4B)
    g1[1] = (int)((ndw & 0xFFFFu) << 16);                      // tensor_dim0[15:0]
    g1[2] = (int)(((ndw >> 16) & 0xFFFFu) | (1u << 16));       // tensor_dim0[31:16], tensor_dim1=1
    g1[3] = (int)((ndw & 0xFFFFu) << 16);                      // tile_dim0 = ndw
    g1[4] = 0;                                                 // tile_dim1=0 (1-D), tile_dim2=0
    g1[5] = (int)ndw;                                          // tensor_dim0_stride (lo)
    g1[6] = 0;
    g1[7] = 0;
    i32x4 z4 = {0, 0, 0, 0};
    i32x8 z8 = {0, 0, 0, 0, 0, 0, 0, 0};
    __builtin_amdgcn_tensor_load_to_lds(g0, g1, z4, z4, z8, 0);
}
#endif

__device__ __forceinline__ uint32_t lds_offset_of(const void* p) {
    // generic pointer to LDS: low 32 bits are the workgroup-relative LDS byte offset
    return (uint32_t)(uintptr_t)p;
}

// ---------------------------------------------------------------------------
// SpMM: Y[rows[e], :] += vals[e] * X[cols[e], :]   (segment-sum via f32 atomics)
// 256 threads/block, 16 lanes per edge (each lane owns one float4 of the row).
// Edge chunks (256 edges) staged into double-buffered LDS by the Tensor Data
// Mover; next chunk's DMA is issued before waiting on the current one, with
// s_wait_tensorcnt(3) exploiting per-wave in-order TDM completion.
// ---------------------------------------------------------------------------
__global__ void lightgcn_spmm(const int* __restrict__ rows, const int* __restrict__ cols,
                              const float* __restrict__ vals,
                              const float* __restrict__ X, float* __restrict__ Y,
                              int nnz) {
    __shared__ int   s_rows[2][256];
    __shared__ int   s_cols[2][256];
    __shared__ float s_vals[2][256];

    const int tid  = threadIdx.x;
    const int sub  = tid & 15;   // which float4 of the 64-float row
    const int eoff = tid >> 4;   // edge sub-index within a 16-edge pass
    const int nchunks = (nnz + 255) >> 8;

    int chunk = blockIdx.x;
    if (chunk >= nchunks) return;
    int buf = 0;

#if HAS_TDM
    // Prime the pipeline: stage this block's first (full) chunk into buffer 0.
    if (tid == 0 && (nnz - (chunk << 8)) >= 256) {
        const int base = chunk << 8;
        tdm_load_1d(lds_offset_of(&s_rows[0][0]), rows + base, 256);
        tdm_load_1d(lds_offset_of(&s_cols[0][0]), cols + base, 256);
        tdm_load_1d(lds_offset_of(&s_vals[0][0]), vals + base, 256);
    }
#endif

    for (; chunk < nchunks; chunk += gridDim.x, buf ^= 1) {
        const int base = chunk << 8;
        const int cnt  = (nnz - base < 256) ? (nnz - base) : 256;

#if HAS_TDM
        if (cnt == 256) {
            if (tid == 0) {  // TDM ignores EXEC: one wave-level DMA per descriptor
                const int nc = chunk + gridDim.x;
                if (nc < nchunks && (nnz - (nc << 8)) >= 256) {
                    const int nb = nc << 8;  // prefetch next chunk into the other buffer
                    tdm_load_1d(lds_offset_of(&s_rows[buf ^ 1][0]), rows + nb, 256);
                    tdm_load_1d(lds_offset_of(&s_cols[buf ^ 1][0]), cols + nb, 256);
                    tdm_load_1d(lds_offset_of(&s_vals[buf ^ 1][0]), vals + nb, 256);
                    // 3 prefetch ops outstanding is OK; oldest 3 (current chunk) must be done
                    __builtin_amdgcn_s_wait_tensorcnt(3);
                } else {
                    __builtin_amdgcn_s_wait_tensorcnt(0);
                }
            }
        } else
#endif
        {
            if (tid < cnt) {  // tail chunk: plain coalesced staging
                s_rows[buf][tid] = rows[base + tid];
                s_cols[buf][tid] = cols[base + tid];
                s_vals[buf][tid] = vals[base + tid];
            }
        }
        __syncthreads();

        for (int k = 0; k < 16; ++k) {
            const int e = (k << 4) + eoff;
            if (e < cnt) {
                const int   r = s_rows[buf][e];
                const int   c = s_cols[buf][e];
                const float v = s_vals[buf][e];
                const float4 xv = *(const float4*)(X + ((size_t)c << 6) + (sub << 2));
                float* yp = Y + ((size_t)r << 6) + (sub << 2);
                __hip_atomic_fetch_add(yp + 0, v * xv.x, __ATOMIC_RELAXED, __HIP_MEMORY_SCOPE_AGENT);
                __hip_atomic_fetch_add(yp + 1, v * xv.y, __ATOMIC_RELAXED, __HIP_MEMORY_SCOPE_AGENT);
                __hip_atomic_fetch_add(yp + 2, v * xv.z, __ATOMIC_RELAXED, __HIP_MEMORY_SCOPE_AGENT);
                __hip_atomic_fetch_add(yp + 3, v * xv.w, __ATOMIC_RELAXED, __HIP_MEMORY_SCOPE_AGENT);
            }
        }
        __syncthreads();  // all waves done with this buffer before it is DMA-overwritten
    }
}

// X = concat(user_emb, item_emb); out = X (layer-0 term); Y = 0 (atomic target)
__global__ void lightgcn_init(const float4* __restrict__ u, const float4* __restrict__ it,
                              float4* __restrict__ X, float4* __restrict__ Y,
                              float4* __restrict__ out, int nu4, int ntot4) {
    const float4 z = make_float4(0.f, 0.f, 0.f, 0.f);
    for (int i = blockIdx.x * blockDim.x + threadIdx.x; i < ntot4;
         i += gridDim.x * blockDim.x) {
        float4 v = (i < nu4) ? u[i] : it[i - nu4];
        X[i]   = v;
        out[i] = v;
        Y[i]   = z;
    }
}

// out += Y; Z = 0  (Z is the retired ping buffer -> becomes next layer's atomic target)
__global__ void lightgcn_accum_zero(float4* __restrict__ out, const float4* __restrict__ Y,
                                    float4* __restrict__ Z, int n4) {
    const float4 z = make_float4(0.f, 0.f, 0.f, 0.f);
    for (int i = blockIdx.x * blockDim.x + threadIdx.x; i < n4;
         i += gridDim.x * blockDim.x) {
        float4 a = out[i];
        float4 b = Y[i];
        a.x += b.x; a.y += b.y; a.z += b.z; a.w += b.w;
        out[i] = a;
        Z[i] = z;
    }
}

// final layer: out = (out + Y) * 0.25
__global__ void lightgcn_accum_scale(float4* __restrict__ out, const float4* __restrict__ Y,
                                     int n4, float scale) {
    for (int i = blockIdx.x * blockDim.x + threadIdx.x; i < n4;
         i += gridDim.x * blockDim.x) {
        float4 a = out[i];
        float4 b = Y[i];
        a.x = (a.x + b.x) * scale;
        a.y = (a.y + b.y) * scale;
        a.z = (a.z + b.z) * scale;
        a.w = (a.w + b.w) * scale;
        out[i] = a;
    }
}

extern "C" void kernel_launch(void* const* d_in, const int* in_sizes, int n_in,
                              void* d_out, int out_size, void* d_ws, size_t ws_size,
                              hipStream_t stream) {
    (void)n_in; (void)out_size; (void)ws_size;
    const float* user_emb = (const float*)d_in[0];
    const float* item_emb = (const float*)d_in[1];
    const int*   rows     = (const int*)d_in[2];
    const int*   cols     = (const int*)d_in[3];
    const float* vals     = (const float*)d_in[4];
    const int    nnz      = in_sizes[2];

    float* out = (float*)d_out;
    const size_t nfloats = (size_t)NTOT * DIM;        // 19.2M floats, 76.8 MB
    float* X = (float*)d_ws;                          // ping
    float* Y = X + nfloats;                           // pong

    const int ntot4 = (int)(nfloats / 4);
    const int nu4   = NUM_USERS * DIM / 4;

    const dim3 blk(256);
    const int egrid = 2048;                           // elementwise grid
    const int nchunks = (nnz + 255) / 256;
    const int sgrid = (nchunks < 4096) ? nchunks : 4096;

    lightgcn_init<<<egrid, blk, 0, stream>>>((const float4*)user_emb,
                                             (const float4*)item_emb,
                                             (float4*)X, (float4*)Y, (float4*)out,
                                             nu4, ntot4);

    for (int layer = 0; layer < NUM_LAYERS; ++layer) {
        lightgcn_spmm<<<sgrid, blk, 0, stream>>>(rows, cols, vals, X, Y, nnz);
        if (layer < NUM_LAYERS - 1) {
            // accumulate this layer and zero the retired buffer for the next layer
            lightgcn_accum_zero<<<egrid, blk, 0, stream>>>((float4*)out, (const float4*)Y,
                                                           (float4*)X, ntot4);
        } else {
            lightgcn_accum_scale<<<egrid, blk, 0, stream>>>((float4*)out, (const float4*)Y,
                                                            ntot4, 0.25f);
        }
        float* t = X; X = Y; Y = t;                   // next layer reads this layer's output
    }
}